// SceneRenderer_63599875719665
// MI455X (gfx1250) — compile-verified
//
#include <hip/hip_runtime.h>

// ---------------------------------------------------------------------------
// 3D Gaussian splat renderer for MI455X (gfx1250, wave32).
//   K1: per-(view,gaussian) preprocess  (cov3d -> conic, SH color, depth key)
//   K2: O(n^2) stable rank sort + scatter into AoS (12 floats / gaussian,
//       slots 9..11 zero -> slot 9 doubles as the "channel >= 3" zero color)
//   K3: per-pixel front-to-back blend; color accumulation done with
//       V_WMMA_F32_16X16X4_F32 (16 pixels x 3 channels, 4 gaussians/step),
//       gaussian params staged to LDS via async global->LDS, cross-half
//       transmittance prefix exchanged with ds_swizzle SWAPX16.
// ---------------------------------------------------------------------------

typedef float v2f __attribute__((ext_vector_type(2)));
typedef float v8f __attribute__((ext_vector_type(8)));

#define SH_C0 0.28209479177387814f
#define SH_C1 0.4886025119029199f

// -------------------------- Kernel 1: preprocess ---------------------------
__global__ void prep_kernel(const float* __restrict__ xyz,
                            const float* __restrict__ feats,
                            const float* __restrict__ scal,
                            const float* __restrict__ rot,
                            const float* __restrict__ opa,
                            const float* __restrict__ c2w_all,
                            const float* __restrict__ intr_all,
                            float* __restrict__ stage,   // [V][n][10]
                            float* __restrict__ zarr,    // [V][n]
                            int n, int V) {
  int tid = blockIdx.x * blockDim.x + threadIdx.x;
  if (tid >= n * V) return;
  int vi = tid / n, i = tid - vi * n;

  // quaternion -> rotation
  float qw = rot[i*4+0], qx = rot[i*4+1], qy = rot[i*4+2], qz = rot[i*4+3];
  float qn = rsqrtf(qw*qw + qx*qx + qy*qy + qz*qz);
  qw *= qn; qx *= qn; qy *= qn; qz *= qn;
  float R[3][3] = {
    {1.f-2.f*(qy*qy+qz*qz), 2.f*(qx*qy-qw*qz),     2.f*(qx*qz+qw*qy)},
    {2.f*(qx*qy+qw*qz),     1.f-2.f*(qx*qx+qz*qz), 2.f*(qy*qz-qw*qx)},
    {2.f*(qx*qz-qw*qy),     2.f*(qy*qz+qw*qx),     1.f-2.f*(qx*qx+qy*qy)}};
  float s[3] = {scal[i*3+0], scal[i*3+1], scal[i*3+2]};
  float Mm[3][3];
  #pragma unroll
  for (int r = 0; r < 3; ++r)
    #pragma unroll
    for (int c = 0; c < 3; ++c) Mm[r][c] = R[r][c] * s[c];
  float C3[3][3];
  #pragma unroll
  for (int r = 0; r < 3; ++r)
    #pragma unroll
    for (int c = 0; c < 3; ++c)
      C3[r][c] = Mm[r][0]*Mm[c][0] + Mm[r][1]*Mm[c][1] + Mm[r][2]*Mm[c][2];

  // rigid inverse of c2w (b == 1)
  const float* cw = c2w_all + (size_t)vi * 16;
  float Rcw[3][3];
  #pragma unroll
  for (int r = 0; r < 3; ++r)
    #pragma unroll
    for (int c = 0; c < 3; ++c) Rcw[r][c] = cw[c*4 + r];
  float tx = cw[3], ty = cw[7], tz = cw[11];
  float tcw[3];
  #pragma unroll
  for (int r = 0; r < 3; ++r)
    tcw[r] = -(Rcw[r][0]*tx + Rcw[r][1]*ty + Rcw[r][2]*tz);

  float X = xyz[i*3+0], Y = xyz[i*3+1], Z = xyz[i*3+2];
  float px = Rcw[0][0]*X + Rcw[0][1]*Y + Rcw[0][2]*Z + tcw[0];
  float py = Rcw[1][0]*X + Rcw[1][1]*Y + Rcw[1][2]*Z + tcw[1];
  float pz = Rcw[2][0]*X + Rcw[2][1]*Y + Rcw[2][2]*Z + tcw[2];
  bool valid = pz > 0.2f;
  float zs = fmaxf(pz, 0.2f);

  const float* In = intr_all + (size_t)vi * 4;
  float fx = In[0], fy = In[1], cx = In[2], cy = In[3];
  float iz = 1.f / zs, iz2 = iz * iz;
  float u = fx * px * iz + cx;
  float v = fy * py * iz + cy;

  // cov_cam = Rcw * C3 * Rcw^T
  float Tm[3][3], Vc[3][3];
  #pragma unroll
  for (int r = 0; r < 3; ++r)
    #pragma unroll
    for (int c = 0; c < 3; ++c)
      Tm[r][c] = Rcw[r][0]*C3[0][c] + Rcw[r][1]*C3[1][c] + Rcw[r][2]*C3[2][c];
  #pragma unroll
  for (int r = 0; r < 3; ++r)
    #pragma unroll
    for (int c = 0; c < 3; ++c)
      Vc[r][c] = Tm[r][0]*Rcw[c][0] + Tm[r][1]*Rcw[c][1] + Tm[r][2]*Rcw[c][2];

  // J rows: (j00, 0, j02), (0, j11, j12)
  float j00 = fx * iz, j02 = -fx * px * iz2;
  float j11 = fy * iz, j12 = -fy * py * iz2;
  float t0_0 = j00*Vc[0][0] + j02*Vc[2][0];
  float t0_1 = j00*Vc[0][1] + j02*Vc[2][1];
  float t0_2 = j00*Vc[0][2] + j02*Vc[2][2];
  float t1_1 = j11*Vc[1][1] + j12*Vc[2][1];
  float t1_2 = j11*Vc[1][2] + j12*Vc[2][2];
  float A2 = t0_0*j00 + t0_2*j02 + 0.3f;
  float B2 = t0_1*j11 + t0_2*j12;
  float C2 = t1_1*j11 + t1_2*j12 + 0.3f;
  float det  = fmaxf(A2*C2 - B2*B2, 1e-6f);
  float idet = 1.f / det;
  float ca =  C2 * idet;
  float cb = -B2 * idet;
  float cc =  A2 * idet;

  // SH degree-1 color
  float dx = X - tx, dy = Y - ty, dz = Z - tz;
  float dn = rsqrtf(dx*dx + dy*dy + dz*dz);
  dx *= dn; dy *= dn; dz *= dn;
  const float* f = feats + (size_t)i * 12;
  float col[3];
  #pragma unroll
  for (int ch = 0; ch < 3; ++ch)
    col[ch] = fmaxf(SH_C0*f[ch] - SH_C1*dy*f[3+ch] + SH_C1*dz*f[6+ch]
                    - SH_C1*dx*f[9+ch] + 0.5f, 0.f);

  float op   = valid ? opa[i] : 0.f;
  float zkey = valid ? pz : __builtin_inff();

  float* st = stage + ((size_t)vi * n + i) * 10;
  st[0] = zkey; st[1] = u; st[2] = v; st[3] = ca; st[4] = cb; st[5] = cc;
  st[6] = op; st[7] = col[0]; st[8] = col[1]; st[9] = col[2];
  zarr[(size_t)vi * n + i] = zkey;
}

// ------------------- Kernel 2: stable rank sort + scatter ------------------
__global__ void rank_kernel(const float* __restrict__ stage,
                            const float* __restrict__ zarr,
                            float* __restrict__ sorted,  // [V][n][12]
                            int n, int V) {
  int tid = blockIdx.x * blockDim.x + threadIdx.x;
  if (tid >= n * V) return;
  int vi = tid / n, i = tid - vi * n;
  const float* zv = zarr + (size_t)vi * n;
  float zi = zv[i];
  int rank = 0;
  for (int j = 0; j < n; ++j) {
    float zj = zv[j];                       // wave-uniform -> s_load
    rank += (zj < zi) || (zj == zi && j < i);
  }
  const float* st = stage + ((size_t)vi * n + i) * 10;
  float* dst = sorted + ((size_t)vi * n + rank) * 12;
  #pragma unroll
  for (int fI = 0; fI < 9; ++fI) dst[fI] = st[fI + 1];
  // slot 9 must be 0: it is read as the color of "channel >= 3" lanes
  dst[9] = 0.f; dst[10] = 0.f; dst[11] = 0.f;
}

// ----------------------- Kernel 3: WMMA blend/render -----------------------
#if defined(__gfx1250__) && __has_builtin(__builtin_amdgcn_global_load_async_to_lds_b128)
#define ASYNC_LDS 1
#else
#define ASYNC_LDS 0
#endif

// exact parameter types of the async builtin (from clang diagnostic):
//   (int __vector_size(16) AS1*, int __vector_size(16) AS3*, imm int, imm int)
typedef int v4i_vs __attribute__((vector_size(16)));
typedef __attribute__((address_space(1))) v4i_vs g_v4i;
typedef __attribute__((address_space(3))) v4i_vs l_v4i;

__device__ __forceinline__ float eval_alpha(const float* q, float gx, float gy) {
  float dx = gx - q[0];
  float dy = gy - q[1];
  float power = -0.5f * (q[2]*dx*dx + q[4]*dy*dy) - q[3]*dx*dy;
  float al = fminf(0.99f, q[5] * __expf(fminf(power, 0.f)));
  al = (power > 0.f) ? 0.f : al;
  al = (al < 0.00392156862745098f) ? 0.f : al;   // 1/255 cutoff
  return al;
}

// branchless lane L <-> lane L^16 exchange (EXEC-neutral; safe around WMMA)
__device__ __forceinline__ float swap16(float x) {
#if __has_builtin(__builtin_amdgcn_ds_swizzle)
  // group-of-32 swizzle: xor_mask=0x10, or_mask=0, and_mask=0x1f -> SWAPX16
  return __int_as_float(__builtin_amdgcn_ds_swizzle(__float_as_int(x), 0x401f));
#else
  return __shfl_xor(x, 16, 32);
#endif
}

extern __shared__ float s_lds[];   // n*12 floats

__global__ __launch_bounds__(256) void render_kernel(
    const float* __restrict__ sorted, float* __restrict__ out,
    const int* __restrict__ Wp, int n, int HW, int bpv) {
  const int W    = *Wp;
  const int view = blockIdx.x / bpv;
  const int lane = threadIdx.x & 31;
  const int wave = threadIdx.x >> 5;
  const int half = lane >> 4;          // 0: gaussians K0,K1 | 1: K2,K3
  const int colN = lane & 15;          // pixel row (A) and channel col (B/C)
  const int csel = (colN < 3) ? colN : 3;   // slot 9 (= 6+3) is a stored 0.0f

  // ---- stage sorted gaussian params into LDS (48KB @ n=1024) ----
  const float* src = sorted + (size_t)view * n * 12;
  const int totalF4 = (n * 12) >> 2;
  for (int t = threadIdx.x; t < totalF4; t += blockDim.x) {
#if ASYNC_LDS
    __builtin_amdgcn_global_load_async_to_lds_b128(
        (g_v4i*)(unsigned long long)(const char*)(src + t * 4),
        (l_v4i*)(unsigned int)(unsigned long long)(const char*)(&s_lds[t * 4]),
        0, 0);
#else
    ((float4*)s_lds)[t] = ((const float4*)src)[t];
#endif
  }
#if ASYNC_LDS
#if __has_builtin(__builtin_amdgcn_s_wait_asynccnt)
  __builtin_amdgcn_s_wait_asynccnt(0);
#else
  asm volatile("s_wait_asynccnt 0" ::: "memory");
#endif
#endif
  __syncthreads();

  // ---- pixel assignment: wave owns 16 consecutive pixels ----
  const int pixbase = (blockIdx.x % bpv) * 128 + wave * 16;
  const int p  = pixbase + colN;
  const int pyi = p / W;
  const float gx = (float)(p - pyi * W) + 0.5f;
  const float gy = (float)pyi + 0.5f;

  float T = 1.f;
  v8f acc = {0.f, 0.f, 0.f, 0.f, 0.f, 0.f, 0.f, 0.f};

  #pragma unroll 2
  for (int c4 = 0; c4 < n; c4 += 4) {
    const float* q0 = &s_lds[(c4 + half * 2) * 12];
    const float* q1 = q0 + 12;
    float a0 = eval_alpha(q0, gx, gy);
    float a1 = eval_alpha(q1, gx, gy);
    float ph = (1.f - a0) * (1.f - a1);          // this half's (1-a) product
    float po = swap16(ph);                       // other half's product
    float pre = half ? po : 1.f;                 // prefix for K2,K3
    float w0 = T * pre * a0;                     // wgt = alpha * T_prev
    float w1 = T * pre * (1.f - a0) * a1;
    T *= ph * po;

    // B: colors; lanes 3..15 read the zero pad slot (branchless, no exec mask)
    float b0 = q0[6 + csel];
    float b1 = q1[6 + csel];

    v2f A = {w0, w1};
    v2f B = {b0, b1};
#if defined(__gfx1250__)
    acc = __builtin_amdgcn_wmma_f32_16x16x4_f32(false, A, false, B,
                                                (short)0, acc, false, false);
#else
    // host-pass / non-gfx1250 placeholder (never executed on device)
    acc[0] += w0 * b0 + w1 * b1;
#endif
  }

  // ---- scatter C: VGPR r -> row M = half*8 + r, channel = colN ----
  if (colN < 3) {
    float* ob = out + ((size_t)view * 3 + colN) * HW;
    int pb = pixbase + half * 8;
    #pragma unroll
    for (int r = 0; r < 8; ++r) ob[pb + r] = acc[r];
  }
}

// ------------------------------- launcher ----------------------------------
extern "C" void kernel_launch(void* const* d_in, const int* in_sizes, int n_in,
                              void* d_out, int out_size, void* d_ws, size_t ws_size,
                              hipStream_t stream) {
  const float* xyz   = (const float*)d_in[0];
  const float* feats = (const float*)d_in[1];
  const float* scal  = (const float*)d_in[2];
  const float* rot   = (const float*)d_in[3];
  const float* opa   = (const float*)d_in[4];
  const int*   Wp    = (const int*)d_in[6];
  const float* c2w   = (const float*)d_in[7];
  const float* intr  = (const float*)d_in[8];

  int n  = in_sizes[0] / 3;          // 1024
  int V  = in_sizes[7] / 16;         // 2 views (b == 1)
  int HW = out_size / (3 * V);       // 16384

  float* stage  = (float*)d_ws;                     // V*n*10 floats
  float* sorted = stage + (size_t)V * n * 10;       // V*n*12 floats (16B aligned)
  float* zarr   = sorted + (size_t)V * n * 12;      // V*n floats

  int total = V * n;
  prep_kernel<<<(total + 255) / 256, 256, 0, stream>>>(
      xyz, feats, scal, rot, opa, c2w, intr, stage, zarr, n, V);
  rank_kernel<<<(total + 255) / 256, 256, 0, stream>>>(
      stage, zarr, sorted, n, V);

  int bpv = HW / 128;                               // 128-pixel blocks per view
  size_t ldsbytes = (size_t)n * 12 * sizeof(float); // 48KB
  render_kernel<<<V * bpv, 256, ldsbytes, stream>>>(
      sorted, (float*)d_out, Wp, n, HW, bpv);
}